// NTKernel_4174708212317
// MI455X (gfx1250) — compile-verified
//
#include <hip/hip_runtime.h>
#include <hip/hip_bf16.h>
#include <cstddef>
#include <cstdint>

typedef __attribute__((ext_vector_type(2))) float v2f;
typedef __attribute__((ext_vector_type(8))) float v8f;
typedef __attribute__((ext_vector_type(4))) unsigned int v4u;
typedef __attribute__((ext_vector_type(4))) int v4i;
typedef __attribute__((ext_vector_type(8))) int v8i;

#define PTOT 114176
#define NSAMP 64

// Feature-block offsets inside each sample's feature row (length PTOT)
#define FW0 0
#define FB0 1728
#define FW1 1792
#define FB1 38656
#define FW2 38720
#define FB2 75584
#define FW3 75648
#define FB3 112512
#define FH  112576

__device__ static inline v8f wmma_f32_16x16x4(v2f a, v2f b, v8f c) {
  // D = A(16x4,f32) x B(4x16,f32) + C(16x16,f32)
  return __builtin_amdgcn_wmma_f32_16x16x4_f32(false, a, false, b, (short)0, c,
                                               false, false);
}

// ---------------------------------------------------------------------------
// Forward conv 3x3 SAME + bias + ReLU, implicit GEMM with WMMA.
// M = 64 out-channels, K = Cin*9, N = H*H spatial.
// grid: (ceil(S/16), 4, NSAMP), block: 32 (one wave per 16x16 tile).
// Branch-free predicated gathers; (ci,kh,kw) tracked incrementally.
// ---------------------------------------------------------------------------
__global__ void conv_fwd_wmma(const float* __restrict__ in,  // [NSAMP][Cin*S]
                              const float* __restrict__ w,   // [64*Cin*9]
                              const float* __restrict__ bias,// [64]
                              float* __restrict__ rout,      // [NSAMP][64*S]
                              int Cin, int H) {
  const int S = H * H;
  const int K = Cin * 9;
  const int lane = threadIdx.x, half = lane >> 4, l15 = lane & 15;
  const int m = blockIdx.y * 16 + l15;
  const int samp = blockIdx.z;
  const float* ain = in + (size_t)samp * Cin * S;
  const float* wrow = w + (size_t)m * K;
  const int n = blockIdx.x * 16 + l15;
  const bool nok = (n < S);
  const int nc = nok ? n : 0;
  const int rh = nc / H;
  const int rw = nc - rh * H;
  int ci[2], kh[2], kw[2];
#pragma unroll
  for (int j = 0; j < 2; ++j) {
    const int k0 = half * 2 + j;          // 0..3 (< 9)
    ci[j] = 0;
    kh[j] = k0 / 3;
    kw[j] = k0 - kh[j] * 3;
  }
  v8f acc = {};
  const int nfull = K >> 2;
  for (int kc = 0; kc < nfull; ++kc) {
    v2f A, B;
#pragma unroll
    for (int j = 0; j < 2; ++j) {
      A[j] = wrow[kc * 4 + half * 2 + j];
      const int ih = rh + kh[j] - 1;
      const int iw = rw + kw[j] - 1;
      const bool ok = nok && ih >= 0 && ih < H && iw >= 0 && iw < H;
      const int idx = ok ? (ci[j] * S + ih * H + iw) : 0;
      const float bv = ain[idx];
      B[j] = ok ? bv : 0.f;
      // advance k by 4: (kh,kw) += (1,1) with carries
      kw[j] += 1; kh[j] += 1;
      if (kw[j] >= 3) { kw[j] -= 3; kh[j] += 1; }
      if (kh[j] >= 3) { kh[j] -= 3; ci[j] += 1; }
    }
    acc = wmma_f32_16x16x4(A, B, acc);
  }
  if (K & 3) {  // tail chunk (layer 1: K=27)
    v2f A, B;
#pragma unroll
    for (int j = 0; j < 2; ++j) {
      const int k = nfull * 4 + half * 2 + j;
      const bool kok = k < K;
      const float av = wrow[kok ? k : 0];
      A[j] = kok ? av : 0.f;
      const int ih = rh + kh[j] - 1;
      const int iw = rw + kw[j] - 1;
      const bool ok = kok && nok && ih >= 0 && ih < H && iw >= 0 && iw < H;
      const int idx = ok ? (ci[j] * S + ih * H + iw) : 0;
      const float bv = ain[idx];
      B[j] = ok ? bv : 0.f;
    }
    acc = wmma_f32_16x16x4(A, B, acc);
  }
  float* rs = rout + (size_t)samp * 64 * S;
  if (nok) {
#pragma unroll
    for (int v = 0; v < 8; ++v) {
      const int mo = blockIdx.y * 16 + half * 8 + v;  // C/D row layout
      const float val = acc[v] + bias[mo];
      rs[(size_t)mo * S + n] = val > 0.f ? val : 0.f;
    }
  }
}

// ---------------------------------------------------------------------------
// Conv backward-data: g_in[i,y,x] = sum_{o,kh,kw} w[o,i,kh,kw]*delta[o,y-kh+1,x-kw+1]
// M = i (64), K = o*9 (576), N = spatial. WMMA implicit GEMM.
// ---------------------------------------------------------------------------
__global__ void conv_bwd_data_wmma(const float* __restrict__ delta, // [NSAMP][64*S]
                                   const float* __restrict__ w,     // [64*64*9]
                                   float* __restrict__ gout,        // [NSAMP][64*S]
                                   int H) {
  const int S = H * H;
  const int lane = threadIdx.x, half = lane >> 4, l15 = lane & 15;
  const int m = blockIdx.y * 16 + l15;   // input channel i
  const int samp = blockIdx.z;
  const float* dl = delta + (size_t)samp * 64 * S;
  const int n = blockIdx.x * 16 + l15;
  const bool nok = (n < S);
  const int nc = nok ? n : 0;
  const int y = nc / H;
  const int x = nc - y * H;
  int oc[2], kh[2], kw[2];
#pragma unroll
  for (int j = 0; j < 2; ++j) {
    const int k0 = half * 2 + j;
    oc[j] = 0;
    kh[j] = k0 / 3;
    kw[j] = k0 - kh[j] * 3;
  }
  v8f acc = {};
  for (int kc = 0; kc < 144; ++kc) {   // K = 576, no tail
    v2f A, B;
#pragma unroll
    for (int j = 0; j < 2; ++j) {
      A[j] = w[(size_t)oc[j] * 576 + m * 9 + kh[j] * 3 + kw[j]];
      const int sh = y - kh[j] + 1;
      const int sw = x - kw[j] + 1;
      const bool ok = nok && sh >= 0 && sh < H && sw >= 0 && sw < H;
      const int idx = ok ? (oc[j] * S + sh * H + sw) : 0;
      const float bv = dl[idx];
      B[j] = ok ? bv : 0.f;
      kw[j] += 1; kh[j] += 1;
      if (kw[j] >= 3) { kw[j] -= 3; kh[j] += 1; }
      if (kh[j] >= 3) { kh[j] -= 3; oc[j] += 1; }
    }
    acc = wmma_f32_16x16x4(A, B, acc);
  }
  float* gs = gout + (size_t)samp * 64 * S;
  if (nok) {
#pragma unroll
    for (int v = 0; v < 8; ++v) {
      const int mo = blockIdx.y * 16 + half * 8 + v;
      gs[(size_t)mo * S + n] = acc[v];
    }
  }
}

// ---------------------------------------------------------------------------
// Per-sample weight-gradient features:
//   G[o, i*9+r] = scale * sum_{hw} delta[o,hw] * ain[i, hw + offset(r)]
// M = o (64), N = Cin*9, K = spatial (chunks of 4). Writes scaled block to F.
// ---------------------------------------------------------------------------
__global__ void gradw_wmma(const float* __restrict__ delta, // [NSAMP][64*S]
                           const float* __restrict__ ain,   // [NSAMP][Cin*S]
                           float* __restrict__ F,           // [NSAMP][PTOT]
                           int Cin, int H, int blockoff,
                           const float* __restrict__ scal, int sidx) {
  const int S = H * H;
  const int Nn = Cin * 9;
  const int lane = threadIdx.x, half = lane >> 4, l15 = lane & 15;
  const int samp = blockIdx.z;
  const float* dl = delta + (size_t)samp * 64 * S;
  const float* as = ain + (size_t)samp * Cin * S;
  const int m = blockIdx.y * 16 + l15;
  const float* drow = dl + (size_t)m * S;
  const int ncol = blockIdx.x * 16 + l15;
  const bool cok = ncol < Nn;
  const int nn = cok ? ncol : 0;
  const int ci = nn / 9;
  const int rr = nn - ci * 9;
  const int kh = rr / 3, kw = rr - kh * 3;
  int y[2], x[2];
#pragma unroll
  for (int j = 0; j < 2; ++j) { y[j] = 0; x[j] = half * 2 + j; }
  v8f acc = {};
  const int nfull = S >> 2;
  for (int kc = 0; kc < nfull; ++kc) {
    v2f A, B;
#pragma unroll
    for (int j = 0; j < 2; ++j) {
      A[j] = drow[kc * 4 + half * 2 + j];
      const int ih = y[j] + kh - 1;
      const int iw = x[j] + kw - 1;
      const bool ok = cok && ih >= 0 && ih < H && iw >= 0 && iw < H;
      const int idx = ok ? (ci * S + ih * H + iw) : 0;
      const float bv = as[idx];
      B[j] = ok ? bv : 0.f;
      x[j] += 4;
      if (x[j] >= H) { x[j] -= H; y[j] += 1; }   // 4 < H always
    }
    acc = wmma_f32_16x16x4(A, B, acc);
  }
  if (S & 3) {  // tail (S = 441)
    v2f A, B;
#pragma unroll
    for (int j = 0; j < 2; ++j) {
      const int k = nfull * 4 + half * 2 + j;
      const bool kok = k < S;
      const float av = drow[kok ? k : 0];
      A[j] = kok ? av : 0.f;
      const int ih = y[j] + kh - 1;
      const int iw = x[j] + kw - 1;
      const bool ok = kok && cok && ih >= 0 && ih < H && iw >= 0 && iw < H;
      const int idx = ok ? (ci * S + ih * H + iw) : 0;
      const float bv = as[idx];
      B[j] = ok ? bv : 0.f;
    }
    acc = wmma_f32_16x16x4(A, B, acc);
  }
  if (cok) {
    const float s = scal[sidx];
    float* fs = F + (size_t)samp * PTOT + blockoff;
#pragma unroll
    for (int v = 0; v < 8; ++v) {
      const int mo = blockIdx.y * 16 + half * 8 + v;
      fs[(size_t)mo * Nn + ncol] = s * acc[v];
    }
  }
}

// ---------------------------------------------------------------------------
// 2x2 max pool (VALID). Optional scale + strided output (used to write the
// fc feature block h directly into F, scaled by s8).
// ---------------------------------------------------------------------------
__global__ void pool2(const float* __restrict__ r, float* __restrict__ out,
                      int H, size_t ostride, int ooff,
                      const float* __restrict__ scale_ptr) {
  const int Hp = H / 2;
  const int tot = 64 * Hp * Hp;
  const int idx = blockIdx.x * blockDim.x + threadIdx.x;
  const int samp = blockIdx.z;
  if (idx >= tot) return;
  const int ch = idx / (Hp * Hp);
  const int rem = idx - ch * Hp * Hp;
  const int ph = rem / Hp, pw = rem - ph * Hp;
  const float* rb = r + (size_t)samp * 64 * H * H + (size_t)ch * H * H;
  const int h0 = 2 * ph, w0 = 2 * pw;
  float m = fmaxf(fmaxf(rb[h0 * H + w0], rb[h0 * H + w0 + 1]),
                  fmaxf(rb[(h0 + 1) * H + w0], rb[(h0 + 1) * H + w0 + 1]));
  const float s = scale_ptr ? scale_ptr[0] : 1.f;
  out[(size_t)samp * ostride + ooff + (size_t)ch * Hp * Hp + ph * Hp + pw] = m * s;
}

// ---------------------------------------------------------------------------
// Backward of (relu -> 2x2 maxpool), in place: r (post-relu, pre-pool) is
// overwritten with delta. One thread owns one 2x2 window (race-free).
// ---------------------------------------------------------------------------
__global__ void unpool_relu_bwd(float* __restrict__ r,
                                const float* __restrict__ g, size_t gstride,
                                const int* __restrict__ cptr, int H) {
  const int Hp = H / 2;
  const int Hw = (H + 1) / 2;
  const int tot = 64 * Hw * Hw;
  const int idx = blockIdx.x * blockDim.x + threadIdx.x;
  const int samp = blockIdx.z;
  if (idx >= tot) return;
  const int ch = idx / (Hw * Hw);
  const int rem = idx - ch * Hw * Hw;
  const int ph = rem / Hw, pw = rem - ph * Hw;
  const int S = H * H;
  float* rb = r + (size_t)samp * 64 * S + (size_t)ch * S;
  const int h0 = 2 * ph, w0 = 2 * pw;
  if (ph < Hp && pw < Hp) {
    const float v00 = rb[h0 * H + w0];
    const float v01 = rb[h0 * H + w0 + 1];
    const float v10 = rb[(h0 + 1) * H + w0];
    const float v11 = rb[(h0 + 1) * H + w0 + 1];
    int am = 0;
    float mv = v00;
    if (v01 > mv) { mv = v01; am = 1; }
    if (v10 > mv) { mv = v10; am = 2; }
    if (v11 > mv) { mv = v11; am = 3; }
    const float* gp = g + gstride * samp + (cptr ? (size_t)cptr[0] * 1600 : 0);
    const float gv = gp[(size_t)ch * Hp * Hp + ph * Hp + pw];
    const float pass = (mv > 0.f) ? gv : 0.f;
    rb[h0 * H + w0] = (am == 0) ? pass : 0.f;
    rb[h0 * H + w0 + 1] = (am == 1) ? pass : 0.f;
    rb[(h0 + 1) * H + w0] = (am == 2) ? pass : 0.f;
    rb[(h0 + 1) * H + w0 + 1] = (am == 3) ? pass : 0.f;
  } else {
#pragma unroll
    for (int dh = 0; dh < 2; ++dh)
#pragma unroll
      for (int dw = 0; dw < 2; ++dw) {
        const int h = h0 + dh, w = w0 + dw;
        if (h < H && w < H) rb[h * H + w] = 0.f;
      }
  }
}

// ---------------------------------------------------------------------------
// Bias-gradient features via Tensor Data Mover:
//   F[samp, boff+o] = s * sum_hw delta[o,hw]
// TDM DMAs a [64 x GB_TW] f32 tile of delta into LDS (one descriptor per
// chunk, issued by wave 0, synchronized with s_wait_tensorcnt + barrier),
// then 64 threads reduce their LDS rows. OOB columns return zero per D# spec
// (tensor_dim0 is set to the remaining row length).
// ---------------------------------------------------------------------------
#define GB_TW 128
__global__ void gradb_feat_tdm(const float* __restrict__ delta,
                               float* __restrict__ F, int S, int boff,
                               const float* __restrict__ scal, int sidx) {
  __shared__ float tile[64 * GB_TW];
  const int samp = blockIdx.x;
  const int t = threadIdx.x;  // 0..63
  const float* base = delta + (size_t)samp * 64 * S;
  float acc = 0.f;
  const int nch = (S + GB_TW - 1) / GB_TW;
  for (int c = 0; c < nch; ++c) {
    const int col0 = c * GB_TW;
    if (t < 32) {  // wave 0 issues the DMA
      const unsigned long long ga = (unsigned long long)(uintptr_t)(base + col0);
      const unsigned int ldsoff = (unsigned int)(uintptr_t)(&tile[0]);
      const unsigned int dim0 = (unsigned int)(S - col0);  // remaining row len
      v4u g0;
      g0.x = 1u;                                   // count=1, user descriptor
      g0.y = ldsoff;                               // lds_addr
      g0.z = (unsigned int)ga;                     // global_addr[31:0]
      g0.w = (unsigned int)((ga >> 32) & 0x01FFFFFFu) | 0x80000000u; // type=2
      v8i g1;
      g1[0] = (int)(2u << 16);                     // data_size = 4 bytes
      g1[1] = (int)((dim0 & 0xFFFFu) << 16);       // tensor_dim0[15:0]
      g1[2] = (int)((dim0 >> 16) | (64u << 16));   // dim0[31:16] | tensor_dim1=64
      g1[3] = (int)((unsigned)GB_TW << 16);        // tile_dim0 = GB_TW
      g1[4] = 64;                                  // tile_dim1 = 64 rows
      g1[5] = S;                                   // tensor_dim0_stride[31:0]
      g1[6] = 0;
      g1[7] = 0;
      v4i z4 = {};
      v8i z8 = {};
      __builtin_amdgcn_tensor_load_to_lds(g0, g1, z4, z4, z8, 0);
      __builtin_amdgcn_s_wait_tensorcnt(0);
    }
    __syncthreads();
    const float* row = tile + t * GB_TW;
    for (int i = 0; i < GB_TW; ++i) acc += row[i];
    __syncthreads();
  }
  F[(size_t)samp * PTOT + boff + t] = acc * scal[sidx];
}

// ---------------------------------------------------------------------------
// Final NTK GEMM: out[n,m] = sum_p F1[n,p]*F2[m,p] + s9^2
// 64x64 output = 4x4 tiles of 16x16, one wave each, K = PTOT in chunks of 4.
// ---------------------------------------------------------------------------
__global__ void ntk_gemm_wmma(const float* __restrict__ F1,
                              const float* __restrict__ F2,
                              float* __restrict__ out,
                              const float* __restrict__ scal) {
  const int lane = threadIdx.x;
  const int half = lane >> 4;
  const int l15 = lane & 15;
  const int mtile = blockIdx.y;
  const int ntile = blockIdx.x;
  const v2f* arow = (const v2f*)(F1 + (size_t)(mtile * 16 + l15) * PTOT);
  const v2f* brow = (const v2f*)(F2 + (size_t)(ntile * 16 + l15) * PTOT);
  v8f acc = {};
  for (int kc = 0; kc < PTOT / 4; ++kc) {
    if ((kc & 63) == 0) {
      __builtin_prefetch((const float*)arow + kc * 4 + 512, 0, 1);
      __builtin_prefetch((const float*)brow + kc * 4 + 512, 0, 1);
    }
    const v2f A = arow[kc * 2 + half];
    const v2f B = brow[kc * 2 + half];
    acc = wmma_f32_16x16x4(A, B, acc);
  }
  const float s9 = scal[9];
  const float add = s9 * s9;
#pragma unroll
  for (int v = 0; v < 8; ++v) {
    const int row = mtile * 16 + half * 8 + v;  // index into x1 batch
    const int col = ntile * 16 + l15;           // index into x2 batch
    out[row * 64 + col] = acc[v] + add;
  }
}

// ---------------------------------------------------------------------------

static inline int cdiv(int a, int b) { return (a + b - 1) / b; }

extern "C" void kernel_launch(void* const* d_in, const int* in_sizes, int n_in,
                              void* d_out, int out_size, void* d_ws,
                              size_t ws_size, hipStream_t stream) {
  (void)in_sizes; (void)n_in; (void)out_size; (void)ws_size;
  const float* x1 = (const float*)d_in[0];
  const float* x2 = (const float*)d_in[1];
  const float* scal = (const float*)d_in[2];
  const int* cptr = (const int*)d_in[3];
  const float* w0 = (const float*)d_in[4];
  const float* b0 = (const float*)d_in[5];
  const float* w1 = (const float*)d_in[6];
  const float* b1 = (const float*)d_in[7];
  const float* w2 = (const float*)d_in[8];
  const float* b2 = (const float*)d_in[9];
  const float* w3 = (const float*)d_in[10];
  const float* b3 = (const float*)d_in[11];
  const float* fcw = (const float*)d_in[12];
  // fc_b (d_in[13]): contributes only s9^2, added in ntk_gemm_wmma.
  float* out = (float*)d_out;
  float* ws = (float*)d_ws;

  // workspace layout (floats)
  const size_t oF1 = 0;
  const size_t oF2 = oF1 + (size_t)NSAMP * PTOT;
  const size_t oR1 = oF2 + (size_t)NSAMP * PTOT;       // 64ch x 84^2 (r -> delta)
  const size_t oR2 = oR1 + (size_t)NSAMP * 64 * 7056;  // 64ch x 42^2
  const size_t oR3 = oR2 + (size_t)NSAMP * 64 * 1764;  // 64ch x 21^2
  const size_t oR4 = oR3 + (size_t)NSAMP * 64 * 441;   // 64ch x 10^2
  const size_t oA1 = oR4 + (size_t)NSAMP * 64 * 100;   // pooled 42^2
  const size_t oA2 = oA1 + (size_t)NSAMP * 64 * 1764;  // pooled 21^2
  const size_t oA3 = oA2 + (size_t)NSAMP * 64 * 441;   // pooled 10^2
  const size_t oG1 = oA3 + (size_t)NSAMP * 64 * 100;   // grad wrt a1
  const size_t oG2 = oG1 + (size_t)NSAMP * 64 * 1764;  // grad wrt a2
  const size_t oG3 = oG2 + (size_t)NSAMP * 64 * 441;   // grad wrt a3

  float* F[2] = {ws + oF1, ws + oF2};
  float* r1 = ws + oR1; float* r2 = ws + oR2; float* r3 = ws + oR3; float* r4 = ws + oR4;
  float* a1 = ws + oA1; float* a2 = ws + oA2; float* a3 = ws + oA3;
  float* g1 = ws + oG1; float* g2 = ws + oG2; float* g3 = ws + oG3;

  const float* xs[2] = {x1, x2};

  for (int bt = 0; bt < 2; ++bt) {
    const float* x = xs[bt];
    float* Fb = F[bt];
    // ---- forward ----
    conv_fwd_wmma<<<dim3(cdiv(7056, 16), 4, NSAMP), 32, 0, stream>>>(x, w0, b0, r1, 3, 84);
    pool2<<<dim3(cdiv(64 * 42 * 42, 256), 1, NSAMP), 256, 0, stream>>>(
        r1, a1, 84, (size_t)64 * 1764, 0, nullptr);
    conv_fwd_wmma<<<dim3(cdiv(1764, 16), 4, NSAMP), 32, 0, stream>>>(a1, w1, b1, r2, 64, 42);
    pool2<<<dim3(cdiv(64 * 21 * 21, 256), 1, NSAMP), 256, 0, stream>>>(
        r2, a2, 42, (size_t)64 * 441, 0, nullptr);
    conv_fwd_wmma<<<dim3(cdiv(441, 16), 4, NSAMP), 32, 0, stream>>>(a2, w2, b2, r3, 64, 21);
    pool2<<<dim3(cdiv(64 * 10 * 10, 256), 1, NSAMP), 256, 0, stream>>>(
        r3, a3, 21, (size_t)64 * 100, 0, nullptr);
    conv_fwd_wmma<<<dim3(cdiv(100, 16), 4, NSAMP), 32, 0, stream>>>(a3, w3, b3, r4, 64, 10);
    // fc feature block h (scaled by s8) straight into F
    pool2<<<dim3(cdiv(64 * 5 * 5, 256), 1, NSAMP), 256, 0, stream>>>(
        r4, Fb, 10, (size_t)PTOT, FH, scal + 8);
    // ---- backward (output index c only) ----
    // layer 4
    unpool_relu_bwd<<<dim3(cdiv(64 * 5 * 5, 256), 1, NSAMP), 256, 0, stream>>>(
        r4, fcw, 0, cptr, 10);
    conv_bwd_data_wmma<<<dim3(cdiv(100, 16), 4, NSAMP), 32, 0, stream>>>(r4, w3, g3, 10);
    gradw_wmma<<<dim3(cdiv(576, 16), 4, NSAMP), 32, 0, stream>>>(r4, a3, Fb, 64, 10, FW3, scal, 6);
    gradb_feat_tdm<<<NSAMP, 64, 0, stream>>>(r4, Fb, 100, FB3, scal, 7);
    // layer 3
    unpool_relu_bwd<<<dim3(cdiv(64 * 11 * 11, 256), 1, NSAMP), 256, 0, stream>>>(
        r3, g3, (size_t)64 * 100, nullptr, 21);
    conv_bwd_data_wmma<<<dim3(cdiv(441, 16), 4, NSAMP), 32, 0, stream>>>(r3, w2, g2, 21);
    gradw_wmma<<<dim3(cdiv(576, 16), 4, NSAMP), 32, 0, stream>>>(r3, a2, Fb, 64, 21, FW2, scal, 4);
    gradb_feat_tdm<<<NSAMP, 64, 0, stream>>>(r3, Fb, 441, FB2, scal, 5);
    // layer 2
    unpool_relu_bwd<<<dim3(cdiv(64 * 21 * 21, 256), 1, NSAMP), 256, 0, stream>>>(
        r2, g2, (size_t)64 * 441, nullptr, 42);
    conv_bwd_data_wmma<<<dim3(cdiv(1764, 16), 4, NSAMP), 32, 0, stream>>>(r2, w1, g1, 42);
    gradw_wmma<<<dim3(cdiv(576, 16), 4, NSAMP), 32, 0, stream>>>(r2, a1, Fb, 64, 42, FW1, scal, 2);
    gradb_feat_tdm<<<NSAMP, 64, 0, stream>>>(r2, Fb, 1764, FB1, scal, 3);
    // layer 1
    unpool_relu_bwd<<<dim3(cdiv(64 * 42 * 42, 256), 1, NSAMP), 256, 0, stream>>>(
        r1, g1, (size_t)64 * 1764, nullptr, 84);
    gradw_wmma<<<dim3(cdiv(27, 16), 4, NSAMP), 32, 0, stream>>>(r1, x, Fb, 3, 84, FW0, scal, 0);
    gradb_feat_tdm<<<NSAMP, 64, 0, stream>>>(r1, Fb, 7056, FB0, scal, 1);
  }

  // K = F1 * F2^T + s9^2
  ntk_gemm_wmma<<<dim3(4, 4), 32, 0, stream>>>(F[0], F[1], out, scal);
}